// Agent_86011015070195
// MI455X (gfx1250) — compile-verified
//
#include <hip/hip_runtime.h>

// ---------------- problem constants ----------------
#define NB 512
#define NT 2048
#define NS 16
#define NH 128
#define NA 4
#define NTOK   (NB*NT)
#define NTILES (NTOK/16)

#define WAVES   2           // waves per block (keeps static LDS < 64KB)
#define TPB     (WAVES*32)
#define NBLOCKS 2048        // 4096 waves -> 16 tiles each

typedef float v2f __attribute__((ext_vector_type(2)));
typedef float v8f __attribute__((ext_vector_type(8)));

__device__ __forceinline__ v8f splat8(float x){
  v8f r;
#pragma unroll
  for (int i = 0; i < 8; i++) r[i] = x;
  return r;
}

__device__ __forceinline__ v2f mk2(float a, float b){
  v2f r; r[0] = a; r[1] = b; return r;
}

// D = A(16x4 f32) * B(4x16 f32) + C(16x16 f32), full fp32 matrix op
__device__ __forceinline__ v8f wmma4(v2f a, v2f b, v8f c){
  return __builtin_amdgcn_wmma_f32_16x16x4_f32(false, a, false, b,
                                               (short)0, c, false, false);
}

// tanh via hw exp2 + rcp; symmetric form avoids inf/NaN for large |x|
__device__ __forceinline__ float fast_tanh(float x){
  float ax = __builtin_fabsf(x);
  float e  = __builtin_amdgcn_exp2f(ax * -2.885390081777927f); // exp(-2|x|) in (0,1]
  float t  = (1.0f - e) * __builtin_amdgcn_rcpf(1.0f + e);
  return __builtin_copysignf(t, x);
}

// within-wave LDS ordering fence: HW keeps DS ops in order per wave; this only
// stops the compiler from reordering the cross-lane store->load sequences.
__device__ __forceinline__ void lds_sync(){
  asm volatile("" ::: "memory");
  __builtin_amdgcn_wave_barrier();
  asm volatile("" ::: "memory");
}

__global__ __launch_bounds__(TPB) void Agent_cbf_qp_kernel(
    const float* __restrict__ state,
    const float* __restrict__ Wc1, const float* __restrict__ bc1,
    const float* __restrict__ Wc2, const float* __restrict__ bc2,
    const float* __restrict__ Wh1, const float* __restrict__ bh1,
    const float* __restrict__ wh2, const float* __restrict__ bh2,
    const float* __restrict__ Wf,  const float* __restrict__ bf,
    const float* __restrict__ Wg,  const float* __restrict__ bg,
    float* __restrict__ out)
{
  // ---- weight B-fragments (per-lane WMMA layout), shared by whole block ----
  __shared__ v2f sBc1 [1024];  // Wc1  16x128 : 8 n-tiles x 4 k-chunks x 32 lanes
  __shared__ v2f sBh1 [1024];  // Wh1  16x128
  __shared__ v2f sBh1T[1024];  // Wh1^T 128x16 : 32 k-chunks x 32 lanes
  __shared__ v2f sBc2 [1024];  // Wc2  128x16 (cols 4..15 zero-padded)
  __shared__ v2f sBf  [ 128];  // Wf   16x16
  __shared__ v2f sBg  [ 512];  // Wg   16x64
  __shared__ float s_bc1[128], s_bh1[128], s_wh2[128], s_bf[16], s_bg[64], s_bc2[4];
  // ---- per-wave scratch ----
  __shared__ float sStage[WAVES][256];   // 16x16 relayout tile
  __shared__ float sDH   [WAVES][256];   // delta_h 16x16
  __shared__ float sFb   [WAVES][256];   // f 16x16
  __shared__ float sGb   [WAVES][1024];  // g 16x64
  __shared__ float sHb   [WAVES][16];    // h
  __shared__ float sUU   [WAVES][64];    // u_unc 16x4

  const int tid = threadIdx.x;

  // ---------- one-time weight permutation into B-fragment layout ----------
  for (int i = tid; i < 1024; i += TPB){
    const int l = i & 31, lm = l & 15, lh = l >> 4;
    const int c = (i >> 5) & 3, jt = i >> 7;        // for K=16 matrices
    const int n = jt*16 + lm,  k0 = c*4 + lh*2;
    sBc1[i] = mk2(Wc1[k0*NH + n], Wc1[(k0+1)*NH + n]);
    sBh1[i] = mk2(Wh1[k0*NH + n], Wh1[(k0+1)*NH + n]);
    const int kc  = i >> 5;                          // for K=128 matrices
    const int kk0 = kc*4 + lh*2;
    sBh1T[i] = mk2(Wh1[lm*NH + kk0], Wh1[lm*NH + kk0 + 1]);   // B[k][n]=Wh1[n][k]
    sBc2[i]  = (lm < NA) ? mk2(Wc2[kk0*NA + lm], Wc2[(kk0+1)*NA + lm])
                         : mk2(0.0f, 0.0f);
  }
  for (int i = tid; i < 128; i += TPB){
    const int l = i & 31, lm = l & 15, lh = l >> 4;
    const int c = i >> 5, k0 = c*4 + lh*2;
    sBf[i] = mk2(Wf[k0*NS + lm], Wf[(k0+1)*NS + lm]);
  }
  for (int i = tid; i < 512; i += TPB){
    const int l = i & 31, lm = l & 15, lh = l >> 4;
    const int c = (i >> 5) & 3, jt = i >> 7;
    const int n = jt*16 + lm, k0 = c*4 + lh*2;
    sBg[i] = mk2(Wg[k0*(NS*NA) + n], Wg[(k0+1)*(NS*NA) + n]);
  }
  for (int i = tid; i < 128; i += TPB){ s_bc1[i]=bc1[i]; s_bh1[i]=bh1[i]; s_wh2[i]=wh2[i]; }
  for (int i = tid; i < 64;  i += TPB) s_bg[i] = bg[i];
  for (int i = tid; i < 16;  i += TPB) s_bf[i] = bf[i];
  for (int i = tid; i < 4;   i += TPB) s_bc2[i] = bc2[i];
  __syncthreads();

  const int lane = tid & 31, wid = tid >> 5;
  const int lm = lane & 15, half = lane >> 4;
  const float bh2s = bh2[0];
  float* stage = sStage[wid];
  float* DHb   = sDH[wid];
  float* Fb    = sFb[wid];
  float* Gb    = sGb[wid];
  float* Hb    = sHb[wid];
  float* UUb   = sUU[wid];

  const int gw = blockIdx.x * WAVES + wid;
  const int stride = gridDim.x * WAVES;

  for (int tile = gw; tile < NTILES; tile += stride){
    // ---- A-matrix: 16 tokens x 16 state dims, A-layout (row=lm, K pairs) ----
    const v2f* sp = (const v2f*)(state + (tile*16 + lm) * NS);
    v2f a[4];
#pragma unroll
    for (int c = 0; c < 4; c++) a[c] = sp[2*c + half];

    if (tile + stride < NTILES)
      __builtin_prefetch(state + ((tile + stride)*16 + lm) * NS, 0, 0);

    // ================= controller: U = tanh(S@Wc1+bc1)@Wc2 + bc2 ============
    v8f U = splat8(lm < NA ? s_bc2[lm] : 0.0f);
    for (int jt = 0; jt < 8; jt++){
      v8f z = splat8(s_bc1[jt*16 + lm]);
#pragma unroll
      for (int c = 0; c < 4; c++)
        z = wmma4(a[c], sBc1[(jt*4 + c)*32 + lane], z);
#pragma unroll
      for (int v = 0; v < 8; v++) z[v] = fast_tanh(z[v]);
      // C-layout (row = v+8*half, col = lm) -> row-major staging
#pragma unroll
      for (int v = 0; v < 8; v++) stage[(v + 8*half)*16 + lm] = z[v];
      lds_sync();
      v2f ta[4];
#pragma unroll
      for (int c = 0; c < 4; c++)
        ta[c] = *(const v2f*)&stage[lm*16 + c*4 + half*2];   // A-layout reload
#pragma unroll
      for (int c = 0; c < 4; c++)
        U = wmma4(ta[c], sBc2[(jt*4 + c)*32 + lane], U);
      lds_sync();
    }

    // ====== CBF: t = tanh(S@Wh1+bh1); h = t.wh2; delta_h = ((1-t^2)wh2)@Wh1^T
    v8f DH = splat8(0.0f);
    float hacc[8];
#pragma unroll
    for (int v = 0; v < 8; v++) hacc[v] = 0.0f;
    for (int jt = 0; jt < 8; jt++){
      v8f z = splat8(s_bh1[jt*16 + lm]);
#pragma unroll
      for (int c = 0; c < 4; c++)
        z = wmma4(a[c], sBh1[(jt*4 + c)*32 + lane], z);
      const float w2 = s_wh2[jt*16 + lm];
#pragma unroll
      for (int v = 0; v < 8; v++){
        const float t = fast_tanh(z[v]);
        hacc[v] += t * w2;
        stage[(v + 8*half)*16 + lm] = (1.0f - t*t) * w2;     // D subtile
      }
      lds_sync();
      v2f ta[4];
#pragma unroll
      for (int c = 0; c < 4; c++)
        ta[c] = *(const v2f*)&stage[lm*16 + c*4 + half*2];
#pragma unroll
      for (int c = 0; c < 4; c++)
        DH = wmma4(ta[c], sBh1T[(jt*4 + c)*32 + lane], DH);
      lds_sync();
    }
    // h: butterfly-reduce over the 16 column-lanes of each half
#pragma unroll
    for (int v = 0; v < 8; v++){
#pragma unroll
      for (int ofs = 1; ofs < 16; ofs <<= 1)
        hacc[v] += __shfl_xor(hacc[v], ofs, 32);
    }
    if (lm == 0){
#pragma unroll
      for (int v = 0; v < 8; v++) Hb[v + 8*half] = hacc[v] + bh2s;
    }

    // ================= dynamics: F = S@Wf + bf, G = S@Wg + bg ===============
    v8f F = splat8(s_bf[lm]);
#pragma unroll
    for (int c = 0; c < 4; c++) F = wmma4(a[c], sBf[c*32 + lane], F);
#pragma unroll
    for (int v = 0; v < 8; v++) Fb[(v + 8*half)*16 + lm] = F[v];

    for (int jt = 0; jt < 4; jt++){
      v8f G = splat8(s_bg[jt*16 + lm]);
#pragma unroll
      for (int c = 0; c < 4; c++) G = wmma4(a[c], sBg[(jt*4 + c)*32 + lane], G);
#pragma unroll
      for (int v = 0; v < 8; v++) Gb[(v + 8*half)*64 + jt*16 + lm] = G[v];
    }

#pragma unroll
    for (int v = 0; v < 8; v++) DHb[(v + 8*half)*16 + lm] = DH[v];
    if (lm < NA){
#pragma unroll
      for (int v = 0; v < 8; v++) UUb[(v + 8*half)*NA + lm] = 2.0f * U[v]; // u_unc
    }
    lds_sync();

    // ================= closed-form CBF-QP, one token per lane ===============
    if (lane < 16){
      const int t = lane;
      float left0 = 0.f, left1 = 0.f, left2 = 0.f, left3 = 0.f;
      float right = Hb[t];
#pragma unroll
      for (int s = 0; s < 16; s++){
        const float dh = DHb[t*16 + s];
        right += dh * Fb[t*16 + s];
        left0 -= dh * Gb[t*64 + s*4 + 0];
        left1 -= dh * Gb[t*64 + s*4 + 1];
        left2 -= dh * Gb[t*64 + s*4 + 2];
        left3 -= dh * Gb[t*64 + s*4 + 3];
      }
      const float uu0 = UUb[t*4+0], uu1 = UUb[t*4+1];
      const float uu2 = UUb[t*4+2], uu3 = UUb[t*4+3];
      const float viol = left0*uu0 + left1*uu1 + left2*uu2 + left3*uu3 - right;
      const float den  = left0*left0 + left1*left1 + left2*left2 + left3*left3 + 1e-8f;
      const float lam  = fmaxf(viol, 0.0f) / den;
      float4 r;
      r.x = uu0 - lam*left0;
      r.y = uu1 - lam*left1;
      r.z = uu2 - lam*left2;
      r.w = uu3 - lam*left3;
      *(float4*)(out + (tile*16 + t)*NA) = r;
    }
    lds_sync();
  }
}

extern "C" void kernel_launch(void* const* d_in, const int* in_sizes, int n_in,
                              void* d_out, int out_size, void* d_ws, size_t ws_size,
                              hipStream_t stream) {
  (void)in_sizes; (void)n_in; (void)d_ws; (void)ws_size; (void)out_size;
  const float* state = (const float*)d_in[0];
  const float* Wc1   = (const float*)d_in[1];
  const float* bc1   = (const float*)d_in[2];
  const float* Wc2   = (const float*)d_in[3];
  const float* bc2   = (const float*)d_in[4];
  const float* Wh1   = (const float*)d_in[5];
  const float* bh1   = (const float*)d_in[6];
  const float* wh2   = (const float*)d_in[7];
  const float* bh2   = (const float*)d_in[8];
  const float* Wf    = (const float*)d_in[9];
  const float* bf    = (const float*)d_in[10];
  const float* Wg    = (const float*)d_in[11];
  const float* bg    = (const float*)d_in[12];
  float* out = (float*)d_out;

  hipLaunchKernelGGL(Agent_cbf_qp_kernel, dim3(NBLOCKS), dim3(TPB), 0, stream,
                     state, Wc1, bc1, Wc2, bc2, Wh1, bh1, wh2, bh2,
                     Wf, bf, Wg, bg, out);
}